// attention3_29446295781874
// MI455X (gfx1250) — compile-verified
//
#include <hip/hip_runtime.h>

typedef float v2f __attribute__((ext_vector_type(2)));
typedef float v4f __attribute__((ext_vector_type(4)));
typedef float v8f __attribute__((ext_vector_type(8)));

#define B_ 4
#define D_ 64
#define N_ 4096   /* 64*64 */
#define KST 20    /* padded LDS row stride (floats): conflict-free for K & V frag reads */

// ---------------------------------------------------------------------------
// Per (b,j): confidence = 1/sum_d exp(q - max_d q)  and  kcol = sum_d kv[b,d,j]
// ---------------------------------------------------------------------------
__global__ void __launch_bounds__(256)
prep_kernel(const float* __restrict__ q, const float* __restrict__ kv,
            float* __restrict__ conf, float* __restrict__ kcol) {
    const int idx = blockIdx.x * blockDim.x + threadIdx.x;   // exact: B*N threads
    const int b = idx >> 12;
    const int j = idx & (N_ - 1);
    const float* pq = q  + (size_t)b * D_ * N_ + j;
    const float* pk = kv + (size_t)b * D_ * N_ + j;
    float m = -__builtin_inff();
#pragma unroll 8
    for (int d = 0; d < D_; ++d) m = fmaxf(m, pq[(size_t)d * N_]);
    float s = 0.f, kc = 0.f;
#pragma unroll 8
    for (int d = 0; d < D_; ++d) {
        s  += __expf(pq[(size_t)d * N_] - m);
        kc += pk[(size_t)d * N_];
    }
    conf[idx] = 1.0f / s;
    kcol[idx] = kc;
}

// ---------------------------------------------------------------------------
// Fused flash attention, 4 waves/block sharing a double-buffered kv tile.
//   S_raw = q^T kv ;  e = a_qk*S + c_k*kcol[j]   (row-const affine terms cancel)
//   tile-wide running max m (wave-uniform, scalar-branch rescale)
//   p = exp(e - m);  P' = p*conf[j]
//   acc  += P' kv^T          (v_wmma_f32_16x16x4_f32)
//   accs += P' * ones        (row-sum of P'  -> bv term)
//   accl += p  * ones        (row-sum of p   -> softmax normalizer)
//   out = gamma*(wv*acc + bv*accs)/accl + kv
// ---------------------------------------------------------------------------
__global__ void __launch_bounds__(128)
flash_attn_kernel(const float* __restrict__ q, const float* __restrict__ kv,
                  const float* __restrict__ conf, const float* __restrict__ kcol,
                  float* __restrict__ out,
                  const float* __restrict__ wq_p, const float* __restrict__ bq_p,
                  const float* __restrict__ wk_p, const float* __restrict__ bk_p,
                  const float* __restrict__ wv_p, const float* __restrict__ bv_p,
                  const float* __restrict__ gamma_p)
{
    const float wq = wq_p[0], bq = bq_p[0], wk = wk_p[0];
    const float wv = wv_p[0], bv = bv_p[0];
    const float gamma = gamma_p[0];
    const float a_qk  = wq * wk;   // scale on raw q.kv dot
    const float c_k   = bq * wk;   // scale on kcol[j] (bk terms are row-const: cancel)

    __shared__ float ktile[2][D_ * KST];   // 2 x 64x16 kv tile (padded) = 10 KB
    __shared__ float Pst[4][16 * 17];      // per-wave P' = p*conf staging
    __shared__ float Plt[4][16 * 17];      // per-wave p staging (normalizer path)

    const int w    = threadIdx.x >> 5;
    const int lane = threadIdx.x & 31;
    const int hi   = lane >> 4;
    const int ln   = lane & 15;

    const int b  = blockIdx.x >> 6;                      // 64 blocks per batch
    const int i0 = ((blockIdx.x & 63) << 6) + (w << 4);  // 64 query rows/block
    const size_t base = (size_t)b * D_ * N_;
    float* Pw = Pst[w];
    float* Pl = Plt[w];

    // Raw-Q A fragments (16x4 f32 layout: lane = M + 16*(K>=2), vgpr = K%2)
    v2f aq[16];
#pragma unroll
    for (int kc = 0; kc < 16; ++kc) {
        const int d = 4 * kc + 2 * hi;
        aq[kc].x = q[base + (size_t)d * N_ + i0 + ln];
        aq[kc].y = q[base + (size_t)(d + 1) * N_ + i0 + ln];
    }

    // Cooperative tile loader: 128 threads x 8 floats = 64x16 tile, coalesced
    const int    ld_d  = threadIdx.x >> 1;
    const int    ld_j  = (threadIdx.x & 1) << 3;
    const float* gsrc  = kv + base + (size_t)ld_d * N_ + ld_j;
    float*       ldst0 = &ktile[0][ld_d * KST + ld_j];
    float*       ldst1 = &ktile[1][ld_d * KST + ld_j];

    {   // preload tile for j0 = 0
        v4f t0 = *(const v4f*)(gsrc);
        v4f t1 = *(const v4f*)(gsrc + 4);
        *(v4f*)(ldst0)     = t0;
        *(v4f*)(ldst0 + 4) = t1;
    }
    __syncthreads();

    v8f acc[4];
#pragma unroll
    for (int t = 0; t < 4; ++t) acc[t] = (v8f){};
    v8f accs = (v8f){};
    v8f accl = (v8f){};
    float m = -__builtin_inff();        // tile-wide running max (wave-uniform)

    const v2f ones = {1.f, 1.f};

    for (int s = 0; s < 256; ++s) {
        const int j0 = s << 4;
        const int jn = (s < 255) ? (j0 + 16) : 0;        // next tile (dummy at end)
        v4f n0 = *(const v4f*)(gsrc + jn);               // prefetch next into regs
        v4f n1 = *(const v4f*)(gsrc + jn + 4);
        __builtin_prefetch(gsrc + ((s < 254) ? (j0 + 32) : 0), 0, 1);

        const float* kt = ktile[s & 1];
        const float cj  = conf[b * N_ + j0 + ln];        // column confidence
        const float ecj = c_k * kcol[b * N_ + j0 + ln];  // column energy bias

        // ---- S = q^T kv (16x16, K=64 as 16 accumulating K=4 WMMAs) ----
        v8f S = (v8f){};
#pragma unroll
        for (int kc = 0; kc < 16; ++kc) {
            const int d = 4 * kc + 2 * hi;
            v2f bkf;
            bkf.x = kt[d * KST + ln];
            bkf.y = kt[(d + 1) * KST + ln];
            S = __builtin_amdgcn_wmma_f32_16x16x4_f32(
                    false, aq[kc], false, bkf, (short)0, S, false, false);
        }

        // ---- energies and tile-wide max (5 butterflies total) ----
        float e[8];
        float mx = -__builtin_inff();
#pragma unroll
        for (int r = 0; r < 8; ++r) {
            e[r] = fmaf(a_qk, S[r], ecj);
            mx = fmaxf(mx, e[r]);
        }
        mx = fmaxf(mx, __shfl_xor(mx, 1, 32));
        mx = fmaxf(mx, __shfl_xor(mx, 2, 32));
        mx = fmaxf(mx, __shfl_xor(mx, 4, 32));
        mx = fmaxf(mx, __shfl_xor(mx, 8, 32));
        mx = fmaxf(mx, __shfl_xor(mx, 16, 32));

        // wave-uniform rescale, taken rarely after warm-up (scalar branch:
        // EXEC stays all-ones for the WMMAs outside)
        if (__builtin_amdgcn_readfirstlane((int)(mx > m))) {
            const float scale = __expf(m - mx);          // 0 on first tile
            m = mx;
#pragma unroll
            for (int r = 0; r < 8; ++r) {
                acc[0][r] *= scale; acc[1][r] *= scale;
                acc[2][r] *= scale; acc[3][r] *= scale;
                accs[r]   *= scale; accl[r]   *= scale;
            }
        }

        // ---- p = exp(e - m); stage p and p*conf (C layout -> LDS) ----
#pragma unroll
        for (int r = 0; r < 8; ++r) {
            const float p = __expf(e[r] - m);
            Pl[(r + 8 * hi) * 17 + ln] = p;
            Pw[(r + 8 * hi) * 17 + ln] = p * cj;
        }

        // ---- acc += P'*kv^T ; accs += P'*ones ; accl += p*ones ----
#pragma unroll
        for (int kc = 0; kc < 4; ++kc) {
            v2f pa, pl;   // A layout: m = ln, k = 4*kc + 2*hi + {0,1}
            pa.x = Pw[ln * 17 + 4 * kc + 2 * hi];
            pa.y = Pw[ln * 17 + 4 * kc + 2 * hi + 1];
            pl.x = Pl[ln * 17 + 4 * kc + 2 * hi];
            pl.y = Pl[ln * 17 + 4 * kc + 2 * hi + 1];
#pragma unroll
            for (int t = 0; t < 4; ++t) {
                const float* vt = kt + (16 * t + ln) * KST + 4 * kc + 2 * hi;
                v2f vbf;
                vbf.x = vt[0];
                vbf.y = vt[1];
                acc[t] = __builtin_amdgcn_wmma_f32_16x16x4_f32(
                             false, pa, false, vbf, (short)0, acc[t], false, false);
            }
            accs = __builtin_amdgcn_wmma_f32_16x16x4_f32(
                       false, pa, false, ones, (short)0, accs, false, false);
            accl = __builtin_amdgcn_wmma_f32_16x16x4_f32(
                       false, pl, false, ones, (short)0, accl, false, false);
        }

        // ---- stage next tile into the other buffer; one barrier per step ----
        float* ldst = (s & 1) ? ldst0 : ldst1;           // buffer (s+1)&1
        *(v4f*)(ldst)     = n0;
        *(v4f*)(ldst + 4) = n1;
        __syncthreads();
    }

    // ---- epilogue: out = gamma*(wv*acc + bv*accs)/accl + kv ----
#pragma unroll
    for (int t = 0; t < 4; ++t) {
        const int dd = 16 * t + ln;
#pragma unroll
        for (int r = 0; r < 8; ++r) {
            const int i = i0 + r + 8 * hi;
            const size_t off = base + (size_t)dd * N_ + i;
            out[off] = gamma * ((wv * acc[t][r] + bv * accs[r]) / accl[r]) + kv[off];
        }
    }
}

// ---------------------------------------------------------------------------
extern "C" void kernel_launch(void* const* d_in, const int* in_sizes, int n_in,
                              void* d_out, int out_size, void* d_ws, size_t ws_size,
                              hipStream_t stream) {
    const float* q  = (const float*)d_in[0];
    const float* kv = (const float*)d_in[1];
    const float* wq = (const float*)d_in[2];
    const float* bq = (const float*)d_in[3];
    const float* wk = (const float*)d_in[4];
    const float* bk = (const float*)d_in[5];
    const float* wv = (const float*)d_in[6];
    const float* bv = (const float*)d_in[7];
    const float* gm = (const float*)d_in[8];
    float* out  = (float*)d_out;
    float* conf = (float*)d_ws;                 // B*N floats
    float* kcol = (float*)d_ws + B_ * N_;       // B*N floats (128 KB total)
    (void)bk;

    prep_kernel<<<(B_ * N_) / 256, 256, 0, stream>>>(q, kv, conf, kcol);
    flash_attn_kernel<<<(B_ * (N_ / 16)) / 4, 128, 0, stream>>>(
        q, kv, conf, kcol, out, wq, bq, wk, bk, wv, bv, gm);
}